// Embed_34643206210175
// MI455X (gfx1250) — compile-verified
//
#include <hip/hip_runtime.h>
#include <cstdint>
#include <cstddef>

#define FEATURES   1024
#define ROW_BYTES  (FEATURES * 4)   /* 4096 bytes per embedding row */

// Plain vector types (HIP's float4 class type is rejected by the builtins).
typedef int   v4i __attribute__((vector_size(16)));
typedef float f4  __attribute__((ext_vector_type(4)));

// ---------------------------------------------------------------------------
// CDNA5 async global<->LDS DMA path (gfx1250): probe for the builtins.
// Signature (from clang diagnostic): (v4i AS1*, v4i AS3*, Ii offset, Ii cpol)
// ---------------------------------------------------------------------------
#if __has_builtin(__builtin_amdgcn_global_load_async_to_lds_b128) && \
    __has_builtin(__builtin_amdgcn_global_store_async_from_lds_b128) && \
    __has_builtin(__builtin_amdgcn_s_wait_asynccnt)
#define HAVE_ASYNC_LDS 1
#else
#define HAVE_ASYNC_LDS 0
#endif

#if HAVE_ASYNC_LDS

typedef __attribute__((address_space(1))) v4i gv4i;   // global v4i
typedef __attribute__((address_space(3))) v4i sv4i;   // LDS v4i

// One wave32 per row. Each lane owns 16 bytes of each 512-byte chunk.
// 8 chunks of 512 B = 4096 B = one full embedding row per wave.
// LDS is a per-wave private 4 KiB bounce buffer -> no barriers needed.
__global__ __launch_bounds__(256) void embed_gather_async(
    const float* __restrict__ emb,
    const int*   __restrict__ idx,
    float*       __restrict__ out,
    int rows)
{
    __shared__ char lds_buf[8 * ROW_BYTES];   // 8 waves * 4 KiB = 32 KiB

    const int tid  = threadIdx.x;
    const int wave = tid >> 5;
    const int lane = tid & 31;
    const int row  = blockIdx.x * 8 + wave;
    if (row >= rows) return;                  // wave-uniform branch

    const int index = idx[row];

    gv4i* gsrc  = (gv4i*)(emb + (size_t)index * FEATURES) + lane;  // lane*16 B
    gv4i* gdst  = (gv4i*)(out + (size_t)row   * FEATURES) + lane;
    sv4i* lbase = (sv4i*)(lds_buf + wave * ROW_BYTES) + lane;

    // 8 async DMA loads in flight (INST_OFFSET applies to both global + LDS addr).
    __builtin_amdgcn_global_load_async_to_lds_b128(gsrc, lbase,    0, 0);
    __builtin_amdgcn_global_load_async_to_lds_b128(gsrc, lbase,  512, 0);
    __builtin_amdgcn_global_load_async_to_lds_b128(gsrc, lbase, 1024, 0);
    __builtin_amdgcn_global_load_async_to_lds_b128(gsrc, lbase, 1536, 0);
    __builtin_amdgcn_global_load_async_to_lds_b128(gsrc, lbase, 2048, 0);
    __builtin_amdgcn_global_load_async_to_lds_b128(gsrc, lbase, 2560, 0);
    __builtin_amdgcn_global_load_async_to_lds_b128(gsrc, lbase, 3072, 0);
    __builtin_amdgcn_global_load_async_to_lds_b128(gsrc, lbase, 3584, 0);

    __builtin_amdgcn_s_wait_asynccnt(0);      // loads landed in LDS

    __builtin_amdgcn_global_store_async_from_lds_b128(gdst, lbase,    0, 0);
    __builtin_amdgcn_global_store_async_from_lds_b128(gdst, lbase,  512, 0);
    __builtin_amdgcn_global_store_async_from_lds_b128(gdst, lbase, 1024, 0);
    __builtin_amdgcn_global_store_async_from_lds_b128(gdst, lbase, 1536, 0);
    __builtin_amdgcn_global_store_async_from_lds_b128(gdst, lbase, 2048, 0);
    __builtin_amdgcn_global_store_async_from_lds_b128(gdst, lbase, 2560, 0);
    __builtin_amdgcn_global_store_async_from_lds_b128(gdst, lbase, 3072, 0);
    __builtin_amdgcn_global_store_async_from_lds_b128(gdst, lbase, 3584, 0);
    // S_ENDPGM performs an implicit wait-idle; stores are guaranteed complete.
}

#endif // HAVE_ASYNC_LDS

// ---------------------------------------------------------------------------
// Fallback / baseline: direct b128 gather with non-temporal stores.
// 256 threads * 16 B = 4096 B = one row per block iteration.
// NT stores keep the 205 MB embedding table resident in the 192 MB L2.
// ---------------------------------------------------------------------------
__global__ __launch_bounds__(256) void embed_gather_direct(
    const f4* __restrict__ emb4,
    const int* __restrict__ idx,
    f4* __restrict__ out4,
    int rows)
{
    const int tid = threadIdx.x;
    for (int row = blockIdx.x; row < rows; row += gridDim.x) {
        const int index = idx[row];
        f4 v = emb4[(size_t)index * (FEATURES / 4) + tid];
        __builtin_nontemporal_store(v, &out4[(size_t)row * (FEATURES / 4) + tid]);
    }
}

extern "C" void kernel_launch(void* const* d_in, const int* in_sizes, int n_in,
                              void* d_out, int out_size, void* d_ws, size_t ws_size,
                              hipStream_t stream)
{
    const float* emb  = (const float*)d_in[0];   // [50257, 1024] fp32
    const int*   idx  = (const int*)d_in[1];     // [4, 4096] int32
    float*       out  = (float*)d_out;           // [4, 4096, 1024] fp32
    const int    rows = in_sizes[1];             // 16384

#if HAVE_ASYNC_LDS
    const int blocks = (rows + 7) / 8;           // 8 waves (rows) per block
    embed_gather_async<<<blocks, 256, 0, stream>>>(emb, idx, out, rows);
#else
    embed_gather_direct<<<rows, 256, 0, stream>>>(
        (const f4*)emb, idx, (f4*)out, rows);
#endif
}